// VGAE_31490700214327
// MI455X (gfx1250) — compile-verified
//
#include <hip/hip_runtime.h>

typedef __attribute__((ext_vector_type(2))) float v2f;
typedef __attribute__((ext_vector_type(8))) float v8f;

#define Nn 50000
#define Ee 800000
#define Bb 8192
#define NF (Nn * 128)   // 6,400,000 floats per [N,128] buffer

// ---------------------------------------------------------------------------
// Generic fp32 WMMA GEMM: C[rows, cols] = A[rows, K] @ Bm[K, cols]
// rows, cols multiples of 16; K multiple of 4. One wave32 per 16x16 tile,
// 4 tiles per 128-thread block. Uses V_WMMA_F32_16X16X4_F32.
// ---------------------------------------------------------------------------
__global__ void gemm_f32_wmma(const float* __restrict__ A,
                              const float* __restrict__ Bm,
                              float* __restrict__ C,
                              int K, int cols, int nTiles) {
    const int tilesN = cols >> 4;
    const int tile = blockIdx.x * 4 + (threadIdx.x >> 5);
    if (tile >= nTiles) return;              // wave-uniform branch: EXEC stays full
    const int tm = (tile / tilesN) << 4;
    const int tn = (tile % tilesN) << 4;
    const int lane = threadIdx.x & 31;
    const int half = lane >> 4;              // 0 or 1
    const int mr   = lane & 15;

    // A element: M = tm+mr, K = k0 + 2*half + {0,1}
    const float* arow = A + (size_t)(tm + mr) * K + 2 * half;
    // B element: K = k0 + 2*half + {0,1}, N = tn+mr
    const float* bcol = Bm + (size_t)(2 * half) * cols + tn + mr;

    v8f acc = {};
    for (int k0 = 0; k0 < K; k0 += 4) {
        v2f a = *(const v2f*)(arow + k0);            // 8B-aligned (even offset)
        v2f b;
        b.x = bcol[(size_t)k0 * cols];
        b.y = bcol[(size_t)(k0 + 1) * cols];
        acc = __builtin_amdgcn_wmma_f32_16x16x4_f32(
            /*neg_a=*/false, a, /*neg_b=*/false, b,
            /*c_mod=*/(short)0, acc, /*reuse_a=*/false, /*reuse_b=*/false);
    }
    // C/D layout: VGPR r holds M = tm + r + 8*half, N = tn + mr
    float* crow = C + (size_t)tm * cols + tn + mr;
#pragma unroll
    for (int r = 0; r < 8; ++r)
        crow[(size_t)(r + 8 * half) * cols] = acc[r];
}

// ---------------------------------------------------------------------------
// adj_logits[B,B] = Z[B,64] @ Z[B,64]^T   (store-bound: 268MB out)
// ---------------------------------------------------------------------------
__global__ void adj_wmma(const float* __restrict__ Z, float* __restrict__ out) {
    const int tilesN = Bb >> 4;                       // 512
    const int tile = blockIdx.x * 4 + (threadIdx.x >> 5);
    const int tm = (tile / tilesN) << 4;
    const int tn = (tile % tilesN) << 4;
    const int lane = threadIdx.x & 31;
    const int half = lane >> 4;
    const int mr   = lane & 15;

    const float* ar = Z + (size_t)(tm + mr) * 64 + 2 * half;   // A: M=tm+mr
    const float* br = Z + (size_t)(tn + mr) * 64 + 2 * half;   // B[k][n]=Z[tn+n][k]

    v8f acc = {};
#pragma unroll
    for (int k0 = 0; k0 < 64; k0 += 4) {
        v2f a = *(const v2f*)(ar + k0);
        v2f b = *(const v2f*)(br + k0);
        acc = __builtin_amdgcn_wmma_f32_16x16x4_f32(
            false, a, false, b, (short)0, acc, false, false);
    }
    float* crow = out + (size_t)tm * Bb + tn + mr;
#pragma unroll
    for (int r = 0; r < 8; ++r)
        crow[(size_t)(r + 8 * half) * Bb] = acc[r];
}

// ---------------------------------------------------------------------------
// Edge scatter: Y[dst, :] += w * X[src, :]  (128 cols; one wave32 per edge,
// each lane covers 4 cols). fp32 atomics resolve in L2 (X,Y are L2-resident).
// ---------------------------------------------------------------------------
__global__ void spmm_scatter(const int* __restrict__ ei,
                             const float* __restrict__ ew,
                             const float* __restrict__ X,
                             float* __restrict__ Y) {
    const int e = blockIdx.x * 8 + (threadIdx.x >> 5);
    if (e >= Ee) return;
    const int lane = threadIdx.x & 31;
    const int src = ei[e];
    const int dst = ei[Ee + e];
    const float w = ew[e];
    const float* xs = X + (size_t)src * 128 + lane;
    float* yd = Y + (size_t)dst * 128 + lane;
#pragma unroll
    for (int i = 0; i < 4; ++i)
        unsafeAtomicAdd(yd + 32 * i, w * xs[32 * i]);
}

__global__ void zero_f32(float* __restrict__ p, int n) {
    int i = blockIdx.x * blockDim.x + threadIdx.x;
    if (i < n) p[i] = 0.0f;
}

// W23[k, 0:64] = W2[k, :], W23[k, 64:128] = W3[k, :]  (128x128)
__global__ void concat_w23(const float* __restrict__ W2,
                           const float* __restrict__ W3,
                           float* __restrict__ W23) {
    int i = blockIdx.x * blockDim.x + threadIdx.x;   // 16384 threads
    int k = i >> 7, j = i & 127;
    W23[i] = (j < 64) ? W2[k * 64 + j] : W3[k * 64 + (j - 64)];
}

// Zb[b, j] = noise[node, j] * exp(relu(S[node, 64+j])) + relu(S[node, j])
__global__ void reparam_gather(const float* __restrict__ S,
                               const float* __restrict__ noise,
                               const int* __restrict__ nodes,
                               float* __restrict__ Zb) {
    int t = blockIdx.x * blockDim.x + threadIdx.x;   // B*64 threads
    int b = t >> 6, j = t & 63;
    int node = nodes[b];
    float m  = S[(size_t)node * 128 + j];
    float ls = S[(size_t)node * 128 + 64 + j];
    m  = m  > 0.0f ? m  : 0.0f;
    ls = ls > 0.0f ? ls : 0.0f;
    Zb[t] = noise[(size_t)node * 64 + j] * __expf(ls) + m;
}

// ---------------------------------------------------------------------------
extern "C" void kernel_launch(void* const* d_in, const int* in_sizes, int n_in,
                              void* d_out, int out_size, void* d_ws, size_t ws_size,
                              hipStream_t stream) {
    const float* features = (const float*)d_in[0];   // [N,256]
    const int*   edge_idx = (const int*)  d_in[1];   // [2,E]
    const float* edge_w   = (const float*)d_in[2];   // [E]
    const float* noise    = (const float*)d_in[3];   // [N,64]
    const int*   nodes    = (const int*)  d_in[4];   // [B]
    const float* W1       = (const float*)d_in[5];   // [256,128]
    const float* W2       = (const float*)d_in[6];   // [128,64]
    const float* W3       = (const float*)d_in[7];   // [128,64]
    float* out = (float*)d_out;                      // [B,B]

    float* buf0 = (float*)d_ws;          // XW1, then HW  [N,128]
    float* buf1 = buf0 + NF;             // hidden, then S [N,128]
    float* w23  = buf1 + NF;             // [128,128]
    float* zb   = w23 + 128 * 128;       // [B,64]

    const int nTiles = (Nn / 16) * (128 / 16);       // 25000, divisible by 4

    // 1) XW1 = features @ W1
    gemm_f32_wmma<<<nTiles / 4, 128, 0, stream>>>(features, W1, buf0, 256, 128, nTiles);
    // 2) hidden = scatter-add of w * XW1[src] into buf1
    zero_f32<<<(NF + 255) / 256, 256, 0, stream>>>(buf1, NF);
    spmm_scatter<<<Ee / 8, 256, 0, stream>>>(edge_idx, edge_w, buf0, buf1);
    // 3) W23 = [W2 | W3]; HW = hidden @ W23 (overwrites buf0)
    concat_w23<<<(128 * 128) / 256, 256, 0, stream>>>(W2, W3, w23);
    gemm_f32_wmma<<<nTiles / 4, 128, 0, stream>>>(buf1, w23, buf0, 128, 128, nTiles);
    // 4) S = scatter-add of w * HW[src] (overwrites buf1 after gemm read completes)
    zero_f32<<<(NF + 255) / 256, 256, 0, stream>>>(buf1, NF);
    spmm_scatter<<<Ee / 8, 256, 0, stream>>>(edge_idx, edge_w, buf0, buf1);
    // 5) reparameterize + gather batch rows
    reparam_gather<<<(Bb * 64) / 256, 256, 0, stream>>>(buf1, noise, nodes, zb);
    // 6) adj_logits = Zb @ Zb^T
    adj_wmma<<<(Bb / 16) * (Bb / 16) / 4, 128, 0, stream>>>(zb, out);
}